// GCNRecommender_6760278524492
// MI455X (gfx1250) — compile-verified
//
#include <hip/hip_runtime.h>
#include <math.h>

typedef __attribute__((ext_vector_type(2))) float v2f;
typedef __attribute__((ext_vector_type(8))) float v8f;

// ---------------------------------------------------------------------------
// WMMA fp32 GEMM: C[N x Fout] = A[N x K] * B[K x Fout] (+ bias)
// block = 128 threads = 4 waves; block tile = 16 rows x 64 cols (16x16/wave)
// ---------------------------------------------------------------------------
template <int K, bool CGUARD>
__global__ __launch_bounds__(128)
void gemm_wmma_f32(const float* __restrict__ A, const float* __restrict__ B,
                   const float* __restrict__ bias, float* __restrict__ C,
                   int Fout) {
  constexpr int KP = K + 1;          // padded LDS stride (bank-conflict free)
  __shared__ float lA[16 * KP];      // full A tile: 16 x K
  __shared__ float lB[64 * 64];      // B chunk: 64 k-rows x 64 cols

  const int tid = threadIdx.x;
  const int wave = tid >> 5;
  const int lane = tid & 31;
  const int half = lane >> 4;        // 0: lanes 0-15, 1: lanes 16-31
  const int l16 = lane & 15;
  const int rowBase = blockIdx.x * 16;
  const int colBase = blockIdx.y * 64;
  const int waveCol = wave * 16;

  // stage A tile (coalesced along K)
  for (int i = tid; i < 16 * K; i += 128) {
    const int r = i / K, k = i - r * K;
    lA[r * KP + k] = A[(size_t)(rowBase + r) * K + k];
  }

  v8f acc = {};
  for (int kc = 0; kc < K; kc += 64) {
    // stage B chunk rows [kc, kc+64)
    for (int i = tid; i < 64 * 64; i += 128) {
      const int kk = i >> 6, c = i & 63;
      const int col = colBase + c;
      float v = 0.0f;
      if (!CGUARD || col < Fout) v = B[(size_t)(kc + kk) * Fout + col];
      lB[i] = v;
    }
    __syncthreads();
#pragma unroll
    for (int kb = 0; kb < 64; kb += 4) {
      const int ka = kc + kb + half * 2;  // A: lanes 0-15 -> K=k,k+1 ; 16-31 -> K=k+2,k+3
      const int kl = kb + half * 2;       // same split for B rows
      v2f a, b;
      a[0] = lA[l16 * KP + ka];
      a[1] = lA[l16 * KP + ka + 1];
      b[0] = lB[kl * 64 + waveCol + l16];
      b[1] = lB[(kl + 1) * 64 + waveCol + l16];
      acc = __builtin_amdgcn_wmma_f32_16x16x4_f32(false, a, false, b,
                                                  (short)0, acc, false, false);
    }
    __syncthreads();
  }

  // D layout: VGPR r -> M = r (lanes 0-15) / r+8 (lanes 16-31), N = l16
#pragma unroll
  for (int r = 0; r < 8; ++r) {
    const int row = rowBase + r + half * 8;
    const int col = colBase + waveCol + l16;
    if (!CGUARD || col < Fout) {
      float v = acc[r];
      if (bias) v += bias[col];
      C[(size_t)row * Fout + col] = v;
    }
  }
}

// ---------------------------------------------------------------------------
// LayerNorm (+ optional ELU), one wave32 per row
// ---------------------------------------------------------------------------
template <int F, bool ELU>
__global__ __launch_bounds__(128)
void ln_elu(const float* __restrict__ in, float* __restrict__ out,
            const float* __restrict__ g, const float* __restrict__ be,
            int Nrows) {
  const int wave = threadIdx.x >> 5, lane = threadIdx.x & 31;
  const int row = blockIdx.x * 4 + wave;
  if (row >= Nrows) return;
  constexpr int PER = F / 32;
  float v[PER];
  float s = 0.0f;
#pragma unroll
  for (int i = 0; i < PER; ++i) {
    v[i] = in[(size_t)row * F + lane + i * 32];
    s += v[i];
  }
#pragma unroll
  for (int o = 16; o > 0; o >>= 1) s += __shfl_xor(s, o, 32);
  const float mu = s * (1.0f / F);
  float q = 0.0f;
#pragma unroll
  for (int i = 0; i < PER; ++i) {
    const float d = v[i] - mu;
    q += d * d;
  }
#pragma unroll
  for (int o = 16; o > 0; o >>= 1) q += __shfl_xor(q, o, 32);
  const float rs = rsqrtf(q * (1.0f / F) + 1e-5f);
#pragma unroll
  for (int i = 0; i < PER; ++i) {
    const int c = lane + i * 32;
    float y = (v[i] - mu) * rs * g[c] + be[c];
    if (ELU) y = y > 0.0f ? y : expm1f(y);
    out[(size_t)row * F + c] = y;
  }
}

// ---------------------------------------------------------------------------
// degree / dinv / coef helpers
// ---------------------------------------------------------------------------
__global__ void zero_f(float* p, int n) {
  int i = blockIdx.x * blockDim.x + threadIdx.x;
  if (i < n) p[i] = 0.0f;
}
__global__ void deg_k(const int* __restrict__ dst, float* __restrict__ deg, int E) {
  int e = blockIdx.x * blockDim.x + threadIdx.x;
  if (e < E)
    __hip_atomic_fetch_add(&deg[dst[e]], 1.0f, __ATOMIC_RELAXED,
                           __HIP_MEMORY_SCOPE_AGENT);
}
__global__ void dinv_k(float* __restrict__ d, int n) {
  int i = blockIdx.x * blockDim.x + threadIdx.x;
  if (i < n) d[i] = rsqrtf(d[i] + 1.0f);  // +1 self loop, always > 0
}
__global__ void coef_k(const int* __restrict__ src, const int* __restrict__ dst,
                       const float* __restrict__ dinv, float* __restrict__ coef,
                       int E) {
  int e = blockIdx.x * blockDim.x + threadIdx.x;
  if (e < E) coef[e] = dinv[src[e]] * dinv[dst[e]];
}

// agg = h * dinv^2 + bias   (self-loop message + bias, full overwrite)
template <int F>
__global__ void self_init(const float* __restrict__ h,
                          const float* __restrict__ dinv,
                          const float* __restrict__ b, float* __restrict__ agg,
                          int Nrows) {
  constexpr int per = F / 4;
  int idx = blockIdx.x * blockDim.x + threadIdx.x;
  if (idx >= Nrows * per) return;
  const int i = idx / per;
  const int c = (idx - i * per) * 4;
  const float d = dinv[i];
  const float d2 = d * d;
  const float4 hv = *(const float4*)(h + (size_t)i * F + c);
  const float4 bv = *(const float4*)(b + c);
  float4 o;
  o.x = hv.x * d2 + bv.x;
  o.y = hv.y * d2 + bv.y;
  o.z = hv.z * d2 + bv.z;
  o.w = hv.w * d2 + bv.w;
  *(float4*)(agg + (size_t)i * F + c) = o;
}

// agg[dst] += h[src] * coef ; consecutive lanes cover consecutive channels
template <int F>
__global__ void scatter_k(const float* __restrict__ h,
                          const float* __restrict__ coef,
                          const int* __restrict__ src,
                          const int* __restrict__ dst, float* __restrict__ agg,
                          int E) {
  constexpr int per = F / 4;
  long long idx = (long long)blockIdx.x * blockDim.x + threadIdx.x;
  if (idx >= (long long)E * per) return;
  const int e = (int)(idx / per);
  const int c = (int)(idx - (long long)e * per) * 4;
  const float cf = coef[e];
  const float4 hv = *(const float4*)(h + (size_t)src[e] * F + c);
  float* p = agg + (size_t)dst[e] * F + c;
  __hip_atomic_fetch_add(p + 0, hv.x * cf, __ATOMIC_RELAXED, __HIP_MEMORY_SCOPE_AGENT);
  __hip_atomic_fetch_add(p + 1, hv.y * cf, __ATOMIC_RELAXED, __HIP_MEMORY_SCOPE_AGENT);
  __hip_atomic_fetch_add(p + 2, hv.z * cf, __ATOMIC_RELAXED, __HIP_MEMORY_SCOPE_AGENT);
  __hip_atomic_fetch_add(p + 3, hv.w * cf, __ATOMIC_RELAXED, __HIP_MEMORY_SCOPE_AGENT);
}

// ---------------------------------------------------------------------------
extern "C" void kernel_launch(void* const* d_in, const int* in_sizes, int n_in,
                              void* d_out, int out_size, void* d_ws,
                              size_t ws_size, hipStream_t stream) {
  const float* x   = (const float*)d_in[0];
  const int*   ei  = (const int*)d_in[1];
  const float* W1  = (const float*)d_in[2];
  const float* b1  = (const float*)d_in[3];
  const float* g1  = (const float*)d_in[4];
  const float* be1 = (const float*)d_in[5];
  const float* W2  = (const float*)d_in[6];
  const float* b2  = (const float*)d_in[7];
  const float* g2  = (const float*)d_in[8];
  const float* be2 = (const float*)d_in[9];
  const float* W3  = (const float*)d_in[10];
  const float* b3  = (const float*)d_in[11];
  const float* g3  = (const float*)d_in[12];
  const float* be3 = (const float*)d_in[13];
  const float* Wl1 = (const float*)d_in[14];
  const float* bl1 = (const float*)d_in[15];
  const float* g4  = (const float*)d_in[16];
  const float* be4 = (const float*)d_in[17];
  const float* Wl2 = (const float*)d_in[18];
  const float* bl2 = (const float*)d_in[19];
  float* out = (float*)d_out;

  const int N = in_sizes[0] / 128;  // 50000 (multiple of 16: 3125 tiles)
  const int E = in_sizes[1] / 2;    // 800000
  const int* src = ei;
  const int* dst = ei + E;

  // workspace layout (floats): dinv[N] | coef[E] | bufA[N*256] | bufB[N*256]
  float* ws = (float*)d_ws;
  float* dinv = ws;
  float* coef = dinv + N;
  float* bufA = coef + E;
  float* bufB = bufA + (size_t)N * 256;

  const int MT = N / 16;  // 3125 row tiles

  // --- degree / normalization coefficients (shared by all 3 GCN layers) ---
  zero_f<<<(N + 255) / 256, 256, 0, stream>>>(dinv, N);
  deg_k<<<(E + 255) / 256, 256, 0, stream>>>(dst, dinv, E);
  dinv_k<<<(N + 255) / 256, 256, 0, stream>>>(dinv, N);
  coef_k<<<(E + 255) / 256, 256, 0, stream>>>(src, dst, dinv, coef, E);

  // --- GCN layer 1: 128 -> 128 ---
  gemm_wmma_f32<128, false><<<dim3(MT, 2), 128, 0, stream>>>(x, W1, nullptr, bufA, 128);
  self_init<128><<<(N * 32 + 255) / 256, 256, 0, stream>>>(bufA, dinv, b1, bufB, N);
  scatter_k<128><<<(int)(((long long)E * 32 + 255) / 256), 256, 0, stream>>>(bufA, coef, src, dst, bufB, E);
  ln_elu<128, true><<<(N + 3) / 4, 128, 0, stream>>>(bufB, bufB, g1, be1, N);

  // --- GCN layer 2: 128 -> 256 ---
  gemm_wmma_f32<128, false><<<dim3(MT, 4), 128, 0, stream>>>(bufB, W2, nullptr, bufA, 256);
  self_init<256><<<(N * 64 + 255) / 256, 256, 0, stream>>>(bufA, dinv, b2, bufB, N);
  scatter_k<256><<<(int)(((long long)E * 64 + 255) / 256), 256, 0, stream>>>(bufA, coef, src, dst, bufB, E);
  ln_elu<256, true><<<(N + 3) / 4, 128, 0, stream>>>(bufB, bufB, g2, be2, N);

  // --- GCN layer 3: 256 -> 128 ---
  gemm_wmma_f32<256, false><<<dim3(MT, 2), 128, 0, stream>>>(bufB, W3, nullptr, bufA, 128);
  self_init<128><<<(N * 32 + 255) / 256, 256, 0, stream>>>(bufA, dinv, b3, bufB, N);
  scatter_k<128><<<(int)(((long long)E * 32 + 255) / 256), 256, 0, stream>>>(bufA, coef, src, dst, bufB, E);
  ln_elu<128, true><<<(N + 3) / 4, 128, 0, stream>>>(bufB, bufB, g3, be3, N);

  // --- lin1: 128 -> 64 (+bias), LN + ELU ---
  gemm_wmma_f32<128, false><<<dim3(MT, 1), 128, 0, stream>>>(bufB, Wl1, bl1, bufA, 64);
  ln_elu<64, true><<<(N + 3) / 4, 128, 0, stream>>>(bufA, bufA, g4, be4, N);

  // --- lin2: 64 -> 500 (+bias), column-guarded ---
  gemm_wmma_f32<64, true><<<dim3(MT, 8), 128, 0, stream>>>(bufA, Wl2, bl2, out, 500);
}